// RNNDecoder_15728170238452
// MI455X (gfx1250) — compile-verified
//
#include <hip/hip_runtime.h>
#include <math.h>

typedef __attribute__((ext_vector_type(2))) float v2f;
typedef __attribute__((ext_vector_type(8))) float v8f;

constexpr int Hh = 1024;   // hidden size
constexpr int Vv = 50257;  // vocab
constexpr int Ll = 12;     // encoder length

// ---------------------------------------------------------------------------
// Wave-level GEMV tile: y[row0..row0+15] += W[rows, 0..K) . sx[0..K)
// via V_WMMA_F32_16X16X4_F32. A = 16x4 tile of W (lane l%16 = row,
// lane>=16 covers K half {2,3}); B = x segment broadcast across the 16
// N columns. Two accumulators break the WMMA C-chain dependency.
// Rows are clamped to [0, nrows) so EXEC stays all-ones (WMMA requirement).
// ---------------------------------------------------------------------------
__device__ __forceinline__ v8f gemv16_acc(const float* __restrict__ W,
                                          const float* __restrict__ sx,
                                          int K, int row0, int nrows, v8f c) {
  const int lane = threadIdx.x & 31;
  const int r    = lane & 15;
  const int kh   = (lane >> 4) << 1;  // 0 for lanes 0-15, 2 for lanes 16-31
  int rc = row0 + r;
  if (rc > nrows - 1) rc = nrows - 1; // clamp: duplicate loads, stores guarded later
  const float* __restrict__ wrow = W + (size_t)rc * (size_t)K + kh;
  const float* __restrict__ xb   = sx + kh;
  v8f c2 = {0.f, 0.f, 0.f, 0.f, 0.f, 0.f, 0.f, 0.f};
#pragma unroll 4
  for (int k = 0; k < K; k += 8) {
    v2f a0 = *(const v2f*)(wrow + k);
    v2f b0 = *(const v2f*)(xb + k);
    v2f a1 = *(const v2f*)(wrow + k + 4);
    v2f b1 = *(const v2f*)(xb + k + 4);
    c  = __builtin_amdgcn_wmma_f32_16x16x4_f32(false, a0, false, b0,
                                               (short)0, c,  false, false);
    c2 = __builtin_amdgcn_wmma_f32_16x16x4_f32(false, a1, false, b1,
                                               (short)0, c2, false, false);
  }
  c += c2;
  return c;
}

// Extract y from D layout (lane 0 -> rows 0..7, lane 16 -> rows 8..15),
// add bias(es), apply activation, store (optionally to two destinations).
template <int ACT>  // 0=none, 1=relu, 2=tanh
__device__ __forceinline__ void store16(v8f c, int row0, int nrows,
                                        const float* __restrict__ b0,
                                        const float* __restrict__ b1,
                                        float* __restrict__ out0,
                                        float* __restrict__ out1) {
  const int lane = threadIdx.x & 31;
  if ((lane & 15) == 0) {
    int base = row0 + ((lane >> 4) << 3);
#pragma unroll
    for (int i = 0; i < 8; ++i) {
      int rr = base + i;
      if (rr < nrows) {
        float y = c[i];
        if (b0) y += b0[rr];
        if (b1) y += b1[rr];
        if (ACT == 1) y = fmaxf(y, 0.f);
        if (ACT == 2) y = tanhf(y);
        out0[rr] = y;
        if (out1) out1[rr] = y;
      }
    }
  }
}

// ---------------------------------------------------------------------------
// K1: embedding + Bahdanau attention + concat.  Tiny; one block.
//   xcat[0:1024)   = emb[tok]
//   xcat[1024:2048)= attn_weights @ encoder_outputs
//   attnw_out[0:12)= softmax scores
// ---------------------------------------------------------------------------
__global__ void k_attn(const int* __restrict__ tok,
                       const float* __restrict__ hidden,
                       const float* __restrict__ enc,
                       const float* __restrict__ emb,
                       const float* __restrict__ attnW,
                       const float* __restrict__ attnb,
                       float* __restrict__ xcat,
                       float* __restrict__ attnw_out) {
  __shared__ float s_in[2 * Hh];
  __shared__ float s_red[256];
  __shared__ float s_w[Ll];
  const int t = threadIdx.x;
  const float* __restrict__ erow = emb + (size_t)tok[0] * Hh;
  for (int i = t; i < Hh; i += 256) {
    s_in[i]      = erow[i];
    s_in[Hh + i] = hidden[i];
  }
  __syncthreads();
  for (int l = 0; l < Ll; ++l) {
    float p = 0.f;
    for (int k = t; k < 2 * Hh; k += 256) p += s_in[k] * attnW[l * 2 * Hh + k];
    s_red[t] = p;
    __syncthreads();
    for (int off = 128; off > 0; off >>= 1) {
      if (t < off) s_red[t] += s_red[t + off];
      __syncthreads();
    }
    if (t == 0) s_w[l] = s_red[0] + attnb[l];
    __syncthreads();
  }
  if (t == 0) {
    float m = s_w[0];
    for (int l = 1; l < Ll; ++l) m = fmaxf(m, s_w[l]);
    float s = 0.f;
    for (int l = 0; l < Ll; ++l) { s_w[l] = expf(s_w[l] - m); s += s_w[l]; }
    float inv = 1.f / s;
    for (int l = 0; l < Ll; ++l) s_w[l] *= inv;
  }
  __syncthreads();
  if (t < Ll) attnw_out[t] = s_w[t];
  for (int h = t; h < Hh; h += 256) {
    float acc = 0.f;
#pragma unroll
    for (int l = 0; l < Ll; ++l) acc += s_w[l] * enc[l * Hh + h];
    xcat[h]      = s_in[h];  // embedded
    xcat[Hh + h] = acc;      // attn_applied
  }
}

// K2: xr = relu(xcat @ comb_W^T + comb_b)   [1024 rows, K=2048]
__global__ void k_comb(const float* __restrict__ combW,
                       const float* __restrict__ combb,
                       const float* __restrict__ xcat,
                       float* __restrict__ xr) {
  __shared__ float sx[2 * Hh];
  for (int i = threadIdx.x; i < 2 * Hh; i += 256) sx[i] = xcat[i];
  __syncthreads();
  const int row0 = blockIdx.x * 128 + (threadIdx.x >> 5) * 16;
  v8f c = {0.f, 0.f, 0.f, 0.f, 0.f, 0.f, 0.f, 0.f};
  c = gemv16_acc(combW, sx, 2 * Hh, row0, Hh, c);
  store16<1>(c, row0, Hh, combb, nullptr, xr, nullptr);
}

// K3: h_new = tanh(xr @ Wih^T + bih + h0 @ Whh^T + bhh)  (fused dual GEMV)
__global__ void k_rnn(const float* __restrict__ Wih,
                      const float* __restrict__ Whh,
                      const float* __restrict__ bih,
                      const float* __restrict__ bhh,
                      const float* __restrict__ xr,
                      const float* __restrict__ h0,
                      float* __restrict__ hnew_ws,
                      float* __restrict__ hnew_out) {
  __shared__ float sx[Hh];
  __shared__ float sh[Hh];
  for (int i = threadIdx.x; i < Hh; i += 256) { sx[i] = xr[i]; sh[i] = h0[i]; }
  __syncthreads();
  const int row0 = blockIdx.x * 128 + (threadIdx.x >> 5) * 16;
  v8f c = {0.f, 0.f, 0.f, 0.f, 0.f, 0.f, 0.f, 0.f};
  c = gemv16_acc(Wih, sx, Hh, row0, Hh, c);
  c = gemv16_acc(Whh, sh, Hh, row0, Hh, c);
  store16<2>(c, row0, Hh, bih, bhh, hnew_ws, hnew_out);
}

// K4: logits = h_new @ out_W^T + out_b    [50257 rows — the 206 MB stream]
__global__ void k_vocab(const float* __restrict__ outW,
                        const float* __restrict__ outb,
                        const float* __restrict__ h,
                        float* __restrict__ logits) {
  __shared__ float sx[Hh];
  for (int i = threadIdx.x; i < Hh; i += 256) sx[i] = h[i];
  __syncthreads();
  const int row0 = blockIdx.x * 128 + (threadIdx.x >> 5) * 16;
  v8f c = {0.f, 0.f, 0.f, 0.f, 0.f, 0.f, 0.f, 0.f};
  c = gemv16_acc(outW, sx, Hh, row0, Vv, c);
  store16<0>(c, row0, Vv, outb, nullptr, logits, nullptr);
}

// K5a/b: global max of logits (two passes through L2-resident data)
__global__ void k_pmax(const float* __restrict__ logits, int n,
                       float* __restrict__ pmax) {
  __shared__ float red[256];
  const int t = threadIdx.x;
  float m = -3.4e38f;
  for (int i = blockIdx.x * 256 + t; i < n; i += gridDim.x * 256)
    m = fmaxf(m, logits[i]);
  red[t] = m;
  __syncthreads();
  for (int off = 128; off > 0; off >>= 1) {
    if (t < off) red[t] = fmaxf(red[t], red[t + off]);
    __syncthreads();
  }
  if (t == 0) pmax[blockIdx.x] = red[0];
}
__global__ void k_maxfin(const float* __restrict__ pmax, int nb,
                         float* __restrict__ M) {
  __shared__ float red[64];
  const int t = threadIdx.x;
  red[t] = (t < nb) ? pmax[t] : -3.4e38f;
  __syncthreads();
  for (int off = 32; off > 0; off >>= 1) {
    if (t < off) red[t] = fmaxf(red[t], red[t + off]);
    __syncthreads();
  }
  if (t == 0) *M = red[0];
}

// K5c/d: sum of exp(logit - M)
__global__ void k_psum(const float* __restrict__ logits, int n,
                       const float* __restrict__ M, float* __restrict__ psum) {
  __shared__ float red[256];
  const float m = *M;
  float s = 0.f;
  for (int i = blockIdx.x * 256 + threadIdx.x; i < n; i += gridDim.x * 256)
    s += expf(logits[i] - m);
  red[threadIdx.x] = s;
  __syncthreads();
  for (int off = 128; off > 0; off >>= 1) {
    if (threadIdx.x < off) red[threadIdx.x] += red[threadIdx.x + off];
    __syncthreads();
  }
  if (threadIdx.x == 0) psum[blockIdx.x] = red[0];
}
__global__ void k_sumfin(const float* __restrict__ psum, int nb,
                         float* __restrict__ logS) {
  __shared__ float red[64];
  const int t = threadIdx.x;
  red[t] = (t < nb) ? psum[t] : 0.f;
  __syncthreads();
  for (int off = 32; off > 0; off >>= 1) {
    if (t < off) red[t] += red[t + off];
    __syncthreads();
  }
  if (t == 0) *logS = logf(red[0]);
}

// K6: logp = logits - M - logS
__global__ void k_final(const float* __restrict__ logits, int n,
                        const float* __restrict__ M,
                        const float* __restrict__ logS,
                        float* __restrict__ logp) {
  const float m = *M, ls = *logS;
  for (int i = blockIdx.x * 256 + threadIdx.x; i < n; i += gridDim.x * 256)
    logp[i] = logits[i] - m - ls;
}

// ---------------------------------------------------------------------------
extern "C" void kernel_launch(void* const* d_in, const int* in_sizes, int n_in,
                              void* d_out, int out_size, void* d_ws,
                              size_t ws_size, hipStream_t stream) {
  (void)in_sizes; (void)n_in; (void)out_size; (void)ws_size;
  const int*   tok    = (const int*)d_in[0];
  const float* hidden = (const float*)d_in[1];
  const float* enc    = (const float*)d_in[2];
  const float* emb    = (const float*)d_in[3];
  const float* attnW  = (const float*)d_in[4];
  const float* attnb  = (const float*)d_in[5];
  const float* combW  = (const float*)d_in[6];
  const float* combb  = (const float*)d_in[7];
  const float* Wih    = (const float*)d_in[8];
  const float* Whh    = (const float*)d_in[9];
  const float* bih    = (const float*)d_in[10];
  const float* bhh    = (const float*)d_in[11];
  const float* outW   = (const float*)d_in[12];
  const float* outb   = (const float*)d_in[13];

  float* out = (float*)d_out;  // [logp (V) | h_new (H) | attn_w (L)]
  float* ws  = (float*)d_ws;
  float* xcat   = ws;                 // 2048
  float* xr     = ws + 2 * Hh;        // 1024
  float* hnew   = ws + 3 * Hh;        // 1024
  float* logits = ws + 4 * Hh;        // V
  float* pmax   = logits + Vv;        // 64
  float* psum   = pmax + 64;          // 64
  float* Mv     = psum + 64;          // 1
  float* lS     = Mv + 1;             // 1

  k_attn<<<1, 256, 0, stream>>>(tok, hidden, enc, emb, attnW, attnb, xcat,
                                out + Vv + Hh);
  k_comb<<<Hh / 128, 256, 0, stream>>>(combW, combb, xcat, xr);
  k_rnn<<<Hh / 128, 256, 0, stream>>>(Wih, Whh, bih, bhh, xr, hidden, hnew,
                                      out + Vv);
  k_vocab<<<(Vv + 127) / 128, 256, 0, stream>>>(outW, outb, hnew, logits);
  k_pmax<<<64, 256, 0, stream>>>(logits, Vv, pmax);
  k_maxfin<<<1, 64, 0, stream>>>(pmax, 64, Mv);
  k_psum<<<64, 256, 0, stream>>>(logits, Vv, Mv, psum);
  k_sumfin<<<1, 64, 0, stream>>>(psum, 64, lS);
  k_final<<<64, 256, 0, stream>>>(logits, Vv, Mv, lS, out);
}